// UnconstrainedFNO_11862699671595
// MI455X (gfx1250) — compile-verified
//
#include <hip/hip_runtime.h>
#include <hip/hip_bf16.h>
#include <math.h>
#include <stdint.h>

// ---------------- types -------------------------------------------------
typedef float v2f  __attribute__((ext_vector_type(2)));
typedef float v8f  __attribute__((ext_vector_type(8)));
typedef unsigned int u32x4 __attribute__((ext_vector_type(4)));
typedef int   i32x4 __attribute__((ext_vector_type(4)));
typedef int   i32x8 __attribute__((ext_vector_type(8)));

#define TWO_PI 6.28318530717958647692f

__device__ __forceinline__ float gelu_exact(float v) {
  return 0.5f * v * (1.0f + erff(v * 0.7071067811865475f));
}
__device__ __forceinline__ float wave_reduce_add(float v) {
#pragma unroll
  for (int off = 16; off > 0; off >>= 1) v += __shfl_down(v, off, 32);
  return v;
}

// ---------------- basis table: 32 x 256, 1/sqrt(W) folded ---------------
__global__ __launch_bounds__(256) void k_basis(float* __restrict__ basis) {
  int w = threadIdx.x;
  for (int c = 0; c < 32; ++c) {
    float val;
    if (c == 0)      val = 1.0f / 16.0f;
    else if (c < 16) val =  cosf(TWO_PI * (float)(c * w) / 256.0f) / 16.0f;
    else if (c == 16) val = 0.0f;                        // irfft drops imag of m=0
    else             val = -sinf(TWO_PI * (float)((c - 16) * w) / 256.0f) / 16.0f;
    basis[c * 256 + w] = val;
  }
}

// ---------------- lift: 3 -> 64 + GELU ----------------------------------
__global__ __launch_bounds__(256) void k_lift(const float* __restrict__ state,
                                              const float* __restrict__ lw,
                                              const float* __restrict__ lb,
                                              float* __restrict__ x) {
  size_t g = (size_t)blockIdx.x * 256 + threadIdx.x;     // over 8*64*65536
  size_t plane = g >> 16;                                // b*64 + o
  int p = (int)(g & 65535);
  int b = (int)(plane >> 6);
  int o = (int)(plane & 63);
  const float* st = state + ((size_t)b * 3) * 65536 + p;
  float v = lb[o] + lw[o*3+0]*st[0] + lw[o*3+1]*st[65536] + lw[o*3+2]*st[131072];
  x[g] = gelu_exact(v);
}

// ---------------- S1: s[b,c,k] = (1/256) sum_{h,w} x * D(h) e^{-i2pi kw/256}
__global__ __launch_bounds__(256) void k_s1(const float* __restrict__ x,
                                            float* __restrict__ s) {
  int bc = blockIdx.x;       // 0..511
  int t  = threadIdx.x;      // w index
  __shared__ float Dre[256], Dim[256];
  __shared__ float accRe[16], accIm[16];
  float dre = 0.f, dim = 0.f;
#pragma unroll
  for (int r = 0; r < 16; ++r) {
    float ang = -TWO_PI * (float)(r * t) / 256.0f;       // e^{-i theta}
    dre += cosf(ang);  dim += sinf(ang);
  }
  Dre[t] = dre; Dim[t] = dim;
  if (t < 16) { accRe[t] = 0.f; accIm[t] = 0.f; }
  __syncthreads();
  const float* xp = x + (size_t)bc * 65536 + t;
  float yre = 0.f, yim = 0.f;
#pragma unroll 4
  for (int h = 0; h < 256; ++h) {
    float v = xp[(size_t)h * 256];
    yre += Dre[h] * v;  yim += Dim[h] * v;
  }
  for (int k = 0; k < 16; ++k) {
    float phi = TWO_PI * (float)(k * t) / 256.0f;
    float cn = cosf(phi), sn = sinf(phi);
    float pr = yre * cn + yim * sn;                      // e^{-i phi} * y
    float pi = yim * cn - yre * sn;
    pr = wave_reduce_add(pr);  pi = wave_reduce_add(pi);
    if ((t & 31) == 0) { atomicAdd(&accRe[k], pr); atomicAdd(&accIm[k], pi); }
  }
  __syncthreads();
  if (t < 16) {
    s[((size_t)bc * 16 + t) * 2 + 0] = accRe[t] * (1.0f / 256.0f);
    s[((size_t)bc * 16 + t) * 2 + 1] = accIm[t] * (1.0f / 256.0f);
  }
}

// ---------------- S2a: blk[b,o',k,m] = sum_i s[b,i,k] * Wc[i,o',k,m] ----
__global__ __launch_bounds__(64) void k_s2a(const float* __restrict__ s,
                                            const float* __restrict__ wr,
                                            const float* __restrict__ wi,
                                            float* __restrict__ blk, int l) {
  int bkm = blockIdx.x;            // b*256 + km
  int b = bkm >> 8, km = bkm & 255, k = km >> 4, m = km & 15;
  int op = threadIdx.x;            // o'
  __shared__ float sr[64], si[64];
  sr[op] = s[(((size_t)b * 64 + op) * 16 + k) * 2 + 0];
  si[op] = s[(((size_t)b * 64 + op) * 16 + k) * 2 + 1];
  __syncthreads();
  float re = 0.f, im = 0.f;
  for (int i = 0; i < 64; ++i) {
    size_t idx = ((((size_t)l * 64 + i) * 64 + op) * 16 + k) * 16 + m;
    float a = wr[idx], bw = wi[idx];
    re += sr[i] * a - si[i] * bw;
    im += sr[i] * bw + si[i] * a;
  }
  size_t o = (((size_t)b * 64 + op) * 256 + km) * 2;
  blk[o] = re;  blk[o + 1] = im;
}

// ---------------- S2b: fold pointwise conv into coefficients ------------
__global__ __launch_bounds__(64) void k_s2b(const float* __restrict__ blk,
                                            const float* __restrict__ pw,
                                            float* __restrict__ blk2, int l) {
  int bkm = blockIdx.x;
  int b = bkm >> 8, km = bkm & 255;
  int o = threadIdx.x;
  __shared__ float cr[64], ci[64];
  size_t base = (((size_t)b * 64 + o) * 256 + km) * 2;
  cr[o] = blk[base];  ci[o] = blk[base + 1];
  __syncthreads();
  const float* pwr = pw + (size_t)l * 4096 + (size_t)o * 64;
  float re = 0.f, im = 0.f;
  for (int op = 0; op < 64; ++op) { float w = pwr[op]; re += w * cr[op]; im += w * ci[op]; }
  blk2[base] = re;  blk2[base + 1] = im;
}

// ---------------- S2g: row-inverse -> Acoef[b,o,h,0:32] (1/sqrt(H) folded)
__global__ __launch_bounds__(256) void k_s2g(const float* __restrict__ blk2,
                                             float* __restrict__ Acoef) {
  int bo = blockIdx.x;       // 0..511
  int h  = threadIdx.x;
  __shared__ float cosT[256], sinT[256], cre[256], cim[256];
  cosT[h] = cosf(TWO_PI * (float)h / 256.0f);
  sinT[h] = sinf(TWO_PI * (float)h / 256.0f);
  cre[h] = blk2[((size_t)bo * 256 + h) * 2 + 0];
  cim[h] = blk2[((size_t)bo * 256 + h) * 2 + 1];
  __syncthreads();
  float* ap = Acoef + ((size_t)bo * 256 + h) * 32;
#pragma unroll
  for (int m = 0; m < 16; ++m) {
    float gre = 0.f, gim = 0.f;
#pragma unroll
    for (int k = 0; k < 16; ++k) {
      int idx = (k * h) & 255;
      float c = cosT[idx], sn = sinT[idx];
      float r = cre[k * 16 + m], ic = cim[k * 16 + m];
      gre += r * c - ic * sn;     // e^{+i 2pi k h/256}
      gim += r * sn + ic * c;
    }
    gre *= (1.0f / 16.0f);  gim *= (1.0f / 16.0f);
    ap[m]      = (m == 0) ? gre : 2.0f * gre;
    ap[16 + m] = 2.0f * gim;      // pairs with -sin basis (row 16 is zero)
  }
}

// ---------------- S3: analytic instance-norm stats (Parseval) -----------
__global__ __launch_bounds__(256) void k_s3(const float* __restrict__ blk2,
                                            float* __restrict__ stats) {
  int bo = blockIdx.x * 256 + threadIdx.x;
  if (bo >= 512) return;
  const float* c = blk2 + (size_t)bo * 512;
  float re00 = c[0];
  float s1 = 0.f, s2 = 0.f;
  for (int k = 1; k < 16; ++k) {
    float r = c[(k * 16) * 2], i = c[(k * 16) * 2 + 1];
    s1 += r * r + i * i;
  }
  for (int m = 1; m < 16; ++m)
    for (int k = 0; k < 16; ++k) {
      float r = c[(k * 16 + m) * 2], i = c[(k * 16 + m) * 2 + 1];
      s2 += r * r + i * i;
    }
  float Ez2  = (re00 * re00 + 0.5f * s1 + 2.0f * s2) * (1.0f / 65536.0f);
  float mean = re00 * (1.0f / 256.0f);                   // bias cancels in IN
  float var  = fmaxf(Ez2 - mean * mean, 0.0f);
  stats[bo * 2 + 0] = mean;
  stats[bo * 2 + 1] = rsqrtf(var + 1e-5f);
}

// ---------------- fused inverse transform + IN + GELU + residual --------
// z[h,w] = Acoef[h,0:32] . basis[0:32,w]   (WMMA f32 16x16x4, K=32)
__global__ __launch_bounds__(256) void k_fuse(const float* __restrict__ Acoef,
                                              const float* __restrict__ basis_g,
                                              const float* __restrict__ stats,
                                              float* __restrict__ x) {
  int blk = blockIdx.x;
  int bo = blk >> 3, hb = blk & 7;
  __shared__ float bs[32 * 256];                          // 32 KB basis tile

  // --- TDM: stage basis (8192 f32, 1-D tile) global -> LDS -------------
  if (threadIdx.x < 32) {
    unsigned long long ga = (unsigned long long)reinterpret_cast<uintptr_t>(basis_g);
    unsigned ldsoff      = (unsigned)reinterpret_cast<uintptr_t>(&bs[0]);
    u32x4 g0;
    g0.x = 1u;                                            // count=1, user mode
    g0.y = ldsoff;                                        // lds_addr
    g0.z = (unsigned)(ga & 0xFFFFFFFFull);                // global_addr[31:0]
    g0.w = (unsigned)((ga >> 32) & 0x1FFFFFFull) | 0x80000000u; // addr[56:32] | type=2
    i32x8 g1 = { (int)0x00020000,        // data_size=4B
                 (int)(8192u << 16),     // tensor_dim0 lo16 = 8192
                 (int)0x00010000,        // tensor_dim0 hi=0 | tensor_dim1 lo16=1
                 (int)(8192u << 16),     // tile_dim0 = 8192 (1-D tile)
                 0,                      // tile_dim1=0, tile_dim2=0
                 8192, 0, 0 };           // tensor_dim0_stride = 8192
    i32x4 z4 = {0, 0, 0, 0};
    i32x8 z8 = {0, 0, 0, 0, 0, 0, 0, 0};
    __builtin_amdgcn_tensor_load_to_lds(g0, g1, z4, z4, z8, 0);
    __builtin_amdgcn_s_wait_tensorcnt(0);
  }
  __syncthreads();

  int tid = threadIdx.x, wid = tid >> 5, lane = tid & 31;
  int htile = wid >> 2, ngrp = wid & 3;
  int h0 = hb * 32 + htile * 16;
  int hl = lane & 15, khalf = lane >> 4;

  // A fragments (8 K-steps of 4) kept in registers, reused across 4 N-tiles
  v2f a[8];
  const float* ap = Acoef + ((size_t)bo * 256 + h0 + hl) * 32 + khalf * 2;
#pragma unroll
  for (int kk = 0; kk < 8; ++kk) a[kk] = *(const v2f*)(ap + kk * 4);

  float mean = stats[bo * 2], istd = stats[bo * 2 + 1];
  size_t base = (size_t)bo * 65536;

#pragma unroll
  for (int j = 0; j < 4; ++j) {
    int w0 = (ngrp * 4 + j) * 16;
    v8f acc = {0.f, 0.f, 0.f, 0.f, 0.f, 0.f, 0.f, 0.f};
#pragma unroll
    for (int kk = 0; kk < 8; ++kk) {
      int row = kk * 4 + khalf * 2;
      v2f bf;
      bf.x = bs[row * 256 + w0 + hl];
      bf.y = bs[(row + 1) * 256 + w0 + hl];
      acc = __builtin_amdgcn_wmma_f32_16x16x4_f32(false, a[kk], false, bf,
                                                  (short)0, acc, false, false);
    }
#pragma unroll
    for (int r = 0; r < 8; ++r) {
      int h = h0 + khalf * 8 + r;
      int w = w0 + hl;
      size_t idx = base + (size_t)h * 256 + w;
      float zn = (acc[r] - mean) * istd;
      x[idx] = gelu_exact(zn) + x[idx];                   // residual, in place
    }
  }
}

// ---------------- out1 (WMMA 64x64 GEMM) + GELU + out2 + update ---------
__global__ __launch_bounds__(256) void k_out(const float* __restrict__ x,
                                             const float* __restrict__ state,
                                             const float* __restrict__ w1,
                                             const float* __restrict__ b1,
                                             const float* __restrict__ w2,
                                             const float* __restrict__ b2,
                                             float* __restrict__ out) {
  int P0 = blockIdx.x * 128;                 // 128 pixels per workgroup
  int b  = P0 >> 16;
  int p0 = P0 & 65535;
  __shared__ float wl[64 * 64];              // out1_w transposed: [c][o]
  __shared__ float hs[128 * 65];             // gelu(out1) tile, padded
  int tid = threadIdx.x;
  for (int idx = tid; idx < 4096; idx += 256) {
    int o = idx >> 6, c = idx & 63;
    wl[c * 64 + o] = w1[o * 64 + c];
  }
  __syncthreads();

  int wid = tid >> 5, lane = tid & 31;
  int pl0 = wid * 16;                        // this wave's 16 pixels
  int ml = lane & 15, kh = lane >> 4;

  v2f a[16];
  const float* xb = x + ((size_t)b * 64) * 65536 + p0 + pl0 + ml;
#pragma unroll
  for (int kk = 0; kk < 16; ++kk) {
    int c0 = kk * 4 + kh * 2;
    a[kk].x = xb[(size_t)c0 * 65536];
    a[kk].y = xb[(size_t)(c0 + 1) * 65536];
  }
#pragma unroll
  for (int j = 0; j < 4; ++j) {
    int n0 = j * 16;
    v8f acc = {0.f, 0.f, 0.f, 0.f, 0.f, 0.f, 0.f, 0.f};
#pragma unroll
    for (int kk = 0; kk < 16; ++kk) {
      int row = kk * 4 + kh * 2;
      v2f bf;
      bf.x = wl[row * 64 + n0 + ml];
      bf.y = wl[(row + 1) * 64 + n0 + ml];
      acc = __builtin_amdgcn_wmma_f32_16x16x4_f32(false, a[kk], false, bf,
                                                  (short)0, acc, false, false);
    }
    int o = n0 + ml;
    float bo1 = b1[o];
#pragma unroll
    for (int r = 0; r < 8; ++r) {
      int pl = pl0 + kh * 8 + r;
      hs[pl * 65 + o] = gelu_exact(acc[r] + bo1);
    }
  }
  __syncthreads();

  if (tid < 128) {
    int pl = tid;
    int P = P0 + pl, bb = P >> 16, pp = P & 65535;
    float d0 = b2[0], d1 = b2[1], d2 = b2[2];
    const float* hp = &hs[pl * 65];
    for (int o = 0; o < 64; ++o) {
      float hv = hp[o];
      d0 += w2[o] * hv;  d1 += w2[64 + o] * hv;  d2 += w2[128 + o] * hv;
    }
    size_t o0 = ((size_t)bb * 3 + 0) * 65536 + pp;
    size_t o1 = o0 + 65536, o2 = o0 + 131072;
    const size_t DT_OFF = 1572864;
    out[DT_OFF + o0] = d0;  out[DT_OFF + o1] = d1;  out[DT_OFF + o2] = d2;
    float s0 = state[o0] + 0.05f * d0;
    float w10 = 10.0f * s0;
    out[o0] = (w10 > 20.0f) ? s0 : log1pf(expf(w10)) * 0.1f;  // softplus beta=10
    out[o1] = state[o1] + 0.05f * d1;
    out[o2] = state[o2] + 0.05f * d2;
  }
}

// ---------------- host launcher -----------------------------------------
extern "C" void kernel_launch(void* const* d_in, const int* in_sizes, int n_in,
                              void* d_out, int out_size, void* d_ws, size_t ws_size,
                              hipStream_t stream) {
  const float* state  = (const float*)d_in[0];
  const float* lift_w = (const float*)d_in[1];
  const float* lift_b = (const float*)d_in[2];
  const float* specWr = (const float*)d_in[3];
  const float* specWi = (const float*)d_in[4];
  const float* pw_w   = (const float*)d_in[5];
  // d_in[6] = pw_b : provably cancelled by affine-free instance norm
  const float* out1_w = (const float*)d_in[7];
  const float* out1_b = (const float*)d_in[8];
  const float* out2_w = (const float*)d_in[9];
  const float* out2_b = (const float*)d_in[10];
  float* out = (float*)d_out;

  float* ws    = (float*)d_ws;
  float* x     = ws;                         // 33,554,432
  float* s     = ws + 33554432;              // 16,384
  float* blk   = ws + 33570816;              // 262,144
  float* blk2  = ws + 33832960;              // 262,144
  float* Acoef = ws + 34095104;              // 4,194,304
  float* basis = ws + 38289408;              // 8,192
  float* stats = ws + 38297600;              // 1,024

  k_basis<<<1, 256, 0, stream>>>(basis);
  k_lift<<<131072, 256, 0, stream>>>(state, lift_w, lift_b, x);
  for (int l = 0; l < 4; ++l) {
    k_s1 <<<512, 256, 0, stream>>>(x, s);
    k_s2a<<<2048, 64, 0, stream>>>(s, specWr, specWi, blk, l);
    k_s2b<<<2048, 64, 0, stream>>>(blk, pw_w, blk2, l);
    k_s2g<<<512, 256, 0, stream>>>(blk2, Acoef);
    k_s3 <<<2, 256, 0, stream>>>(blk2, stats);
    k_fuse<<<4096, 256, 0, stream>>>(Acoef, basis, stats, x);
  }
  k_out<<<4096, 256, 0, stream>>>(x, state, out1_w, out1_b, out2_w, out2_b, out);
  (void)in_sizes; (void)n_in; (void)out_size; (void)ws_size;
}